// AttentionLayer_48722109006175
// MI455X (gfx1250) — compile-verified
//
#include <hip/hip_runtime.h>

typedef float v2f __attribute__((ext_vector_type(2)));
typedef float v8f __attribute__((ext_vector_type(8)));

#define BDIM 8
#define NDIM 1024
#define FDIM 128
#define CDIM 128
#define EFDIM 64
#define HDIM 4
#define NEG_SLOPE 0.2f

// ---------------------------------------------------------------------------
// Kernel 1: fold the three small chains into M_pad[128][16]:
//   cols 0-3 : W_lin @ a_src   (attn_src projection)
//   cols 4-7 : W_lin @ a_dst   (attn_dst projection)
//   cols 8-11: W_edge @ a_edge (edge projection)
//   cols 12-15: zero padding (WMMA N=16)
// ---------------------------------------------------------------------------
__global__ void k_prep(const float* __restrict__ W_lin, const float* __restrict__ a_src,
                       const float* __restrict__ a_dst, const float* __restrict__ W_edge,
                       const float* __restrict__ a_edge, float* __restrict__ Mpad) {
    int t = threadIdx.x;            // 512 threads: (f,h)
    int f = t >> 2, h = t & 3;
    float ms = 0.f, md = 0.f, me = 0.f;
    for (int c = 0; c < CDIM; ++c) {
        float w = W_lin[f * CDIM + c];
        ms += w * a_src[c * HDIM + h];
        md += w * a_dst[c * HDIM + h];
    }
    for (int d = 0; d < EFDIM; ++d)
        me += W_edge[f * EFDIM + d] * a_edge[d * HDIM + h];
    Mpad[f * 16 + h]      = ms;
    Mpad[f * 16 + 4 + h]  = md;
    Mpad[f * 16 + 8 + h]  = me;
    Mpad[f * 16 + 12 + h] = 0.f;
}

// ---------------------------------------------------------------------------
// Kernel 2: P[8192][16] = src_flat[8192][128] @ M_pad[128][16] using
// V_WMMA_F32_16X16X4_F32. One wave32 per 16-row tile; K=128 in steps of 4.
// A layout (16x4 f32): lanes 0-15 hold {K,K+1} of row M=lane,
//                      lanes 16-31 hold {K+2,K+3} of row M=lane-16.
// B layout (4x16 f32): mirrors K split; VGPR0/1 hold rows K+2*hi, K+1+2*hi
//                      at column N = lane&15.
// D layout: VGPR r -> row rowbase + r + 8*hi, col = lane&15.
// ---------------------------------------------------------------------------
__global__ __launch_bounds__(256) void k_proj(const float* __restrict__ src,
                                              const float* __restrict__ Mpad,
                                              float* __restrict__ P) {
    int lane = threadIdx.x & 31;
    int wave = threadIdx.x >> 5;
    int rowbase = blockIdx.x * 128 + wave * 16;   // 64 blocks * 8 waves * 16 rows = 8192
    int hi = lane >> 4;                            // 0 or 1 (K-half select)
    int r  = lane & 15;

    const float* arow = src + (size_t)(rowbase + r) * FDIM;
    v8f acc = {};
    #pragma unroll 4
    for (int k = 0; k < FDIM; k += 4) {
        v2f a, b;
        const float* ap = arow + k + 2 * hi;
        a.x = ap[0];
        a.y = ap[1];
        b.x = Mpad[(k + 2 * hi) * 16 + r];
        b.y = Mpad[(k + 1 + 2 * hi) * 16 + r];
        acc = __builtin_amdgcn_wmma_f32_16x16x4_f32(
            /*neg_a=*/false, a, /*neg_b=*/false, b,
            /*c_mod=*/(short)0, acc, /*reuse_a=*/false, /*reuse_b=*/false);
    }
    #pragma unroll
    for (int v = 0; v < 8; ++v) {
        int row = rowbase + v + 8 * hi;
        P[row * 16 + r] = acc[v];
    }
}

// ---------------------------------------------------------------------------
// CSR build over edges (batch-independent): counts -> rowptr -> col list
// ---------------------------------------------------------------------------
__global__ void k_zero(int* __restrict__ counts) {
    counts[blockIdx.x * 256 + threadIdx.x] = 0;
}

__global__ void k_count(const int* __restrict__ ei0, int* __restrict__ counts, int E) {
    int e = blockIdx.x * 256 + threadIdx.x;
    if (e < E) atomicAdd(&counts[ei0[e]], 1);
}

__global__ void k_scan(const int* __restrict__ counts, int* __restrict__ rowptr,
                       int* __restrict__ cursor) {
    int s = 0;
    for (int i = 0; i < NDIM; ++i) { rowptr[i] = s; cursor[i] = s; s += counts[i]; }
    rowptr[NDIM] = s;
}

__global__ void k_fill(const int* __restrict__ ei0, const int* __restrict__ ei1,
                       int* __restrict__ cursor, int* __restrict__ col, int E) {
    int e = blockIdx.x * 256 + threadIdx.x;
    if (e < E) {
        int i = ei0[e];
        int pos = atomicAdd(&cursor[i], 1);
        col[pos] = ei1[e];
    }
}

// ---------------------------------------------------------------------------
// Kernel 7: streaming output pass. Block = (i = blockIdx.x, b = blockIdx.y).
// LDS holds the sparse edge contribution for this (b,i) row (16 KB).
// Each block writes one contiguous 16 KB slice of d_out exactly once.
// ---------------------------------------------------------------------------
__global__ __launch_bounds__(256) void k_main(const float* __restrict__ P,
                                              const int* __restrict__ rowptr,
                                              const int* __restrict__ col,
                                              float* __restrict__ out) {
    __shared__ __align__(16) float edgerow[NDIM * HDIM];   // 16 KB
    const int i = blockIdx.x, b = blockIdx.y;
    const int tid = threadIdx.x;

    float4* er4 = (float4*)edgerow;
    for (int j = tid; j < NDIM; j += 256) er4[j] = make_float4(0.f, 0.f, 0.f, 0.f);
    __syncthreads();

    const float* Pb = P + (size_t)b * NDIM * 16;
    const float4 pei = *(const float4*)(Pb + i * 16 + 8);   // P_edge[b,i,:]

    const int es = rowptr[i], ee = rowptr[i + 1];
    for (int e = es + tid; e < ee; e += 256) {
        int j = col[e];
        const float* pj = Pb + j * 16 + 8;                  // P_edge[b,j,:]
        atomicAdd(&edgerow[j * 4 + 0], pei.x - pj[0]);
        atomicAdd(&edgerow[j * 4 + 1], pei.y - pj[1]);
        atomicAdd(&edgerow[j * 4 + 2], pei.z - pj[2]);
        atomicAdd(&edgerow[j * 4 + 3], pei.w - pj[3]);
    }
    __syncthreads();

    const float4 ps = *(const float4*)(Pb + i * 16);        // attn_src[b,i,:]
    float* orow = out + ((size_t)(b * NDIM + i)) * NDIM * HDIM;

    for (int j = tid; j < NDIM; j += 256) {
        float4 pd = *(const float4*)(Pb + j * 16 + 4);      // attn_dst[b,j,:]
        float4 er = er4[j];
        float v0 = ps.x + pd.x + er.x;
        float v1 = ps.y + pd.y + er.y;
        float v2 = ps.z + pd.z + er.z;
        float v3 = ps.w + pd.w + er.w;
        // leaky relu
        v0 = (v0 >= 0.f) ? v0 : NEG_SLOPE * v0;
        v1 = (v1 >= 0.f) ? v1 : NEG_SLOPE * v1;
        v2 = (v2 >= 0.f) ? v2 : NEG_SLOPE * v2;
        v3 = (v3 >= 0.f) ? v3 : NEG_SLOPE * v3;
        // softmax over the 4 heads
        float m = fmaxf(fmaxf(v0, v1), fmaxf(v2, v3));
        float e0 = __expf(v0 - m), e1 = __expf(v1 - m);
        float e2 = __expf(v2 - m), e3 = __expf(v3 - m);
        float inv = 1.f / (e0 + e1 + e2 + e3);
        *(float4*)(orow + j * 4) = make_float4(e0 * inv, e1 * inv, e2 * inv, e3 * inv);
    }
}

// ---------------------------------------------------------------------------
extern "C" void kernel_launch(void* const* d_in, const int* in_sizes, int n_in,
                              void* d_out, int out_size, void* d_ws, size_t ws_size,
                              hipStream_t stream) {
    const float* src    = (const float*)d_in[0];   // [8,1024,128]
    const int*   eidx   = (const int*)d_in[1];     // [2,E]
    const float* W_lin  = (const float*)d_in[2];   // [128,128]
    const float* a_src  = (const float*)d_in[3];   // [128,4]
    const float* a_dst  = (const float*)d_in[4];   // [128,4]
    const float* W_edge = (const float*)d_in[5];   // [128,64]
    const float* a_edge = (const float*)d_in[6];   // [64,4]
    float* out = (float*)d_out;                    // [8,1024,1024,4]

    const int E = in_sizes[1] / 2;
    const int* ei0 = eidx;
    const int* ei1 = eidx + E;

    // workspace layout (~656 KB)
    float* Mpad   = (float*)d_ws;                          // 2048 floats
    float* P      = Mpad + 2048;                           // 8192*16 floats
    int*   rowptr = (int*)(P + (size_t)BDIM * NDIM * 16);  // 1025 (+pad)
    int*   counts = rowptr + 1028;                         // 1024
    int*   cursor = counts + NDIM;                         // 1024
    int*   col    = cursor + NDIM;                         // E ints

    k_prep<<<1, 512, 0, stream>>>(W_lin, a_src, a_dst, W_edge, a_edge, Mpad);
    k_proj<<<64, 256, 0, stream>>>(src, Mpad, P);
    k_zero<<<NDIM / 256, 256, 0, stream>>>(counts);
    k_count<<<(E + 255) / 256, 256, 0, stream>>>(ei0, counts, E);
    k_scan<<<1, 1, 0, stream>>>(counts, rowptr, cursor);
    k_fill<<<(E + 255) / 256, 256, 0, stream>>>(ei0, ei1, cursor, col, E);
    k_main<<<dim3(NDIM, BDIM), 256, 0, stream>>>(P, rowptr, col, out);
}